// ModelPBE_23407571763667
// MI455X (gfx1250) — compile-verified
//
#include <hip/hip_runtime.h>
#include <hip/hip_bf16.h>
#include <math.h>
#include <stdint.h>

#define N_MAX 3000
#define M_MAX 400
#define TPB   1024
#define EPT   3            // ceil(3000/1024)

// Discretization constants: K=1.5s, H=0.1um
#define KH   15.0f                      // K/H
#define K2H  7.5f                       // K/(2H)
#define DL   (300.0f / 2999.0f)         // linspace(0,300,3000) spacing

__device__ __forceinline__ unsigned lds_off(const void* p) {
    // Generic pointers into LDS carry the wave-relative LDS byte offset in the
    // low 32 bits (ISA 10.2).
    return (unsigned)(uintptr_t)p;
}

__device__ __forceinline__ void async_load_b128(unsigned lds, const void* gaddr) {
    unsigned long long ga = (unsigned long long)(uintptr_t)gaddr;
    asm volatile("global_load_async_to_lds_b128 %0, %1, off"
                 :: "v"(lds), "v"(ga) : "memory");
}

__device__ __forceinline__ void async_store_b128(void* gaddr, unsigned lds) {
    unsigned long long ga = (unsigned long long)(uintptr_t)gaddr;
    asm volatile("global_store_async_from_lds_b128 %0, %1, off"
                 :: "v"(ga), "v"(lds) : "memory");
}

__device__ __forceinline__ void wait_async0() {
    asm volatile("s_wait_asynccnt 0" ::: "memory");
}

__device__ __forceinline__ float limiter(float fnm1, float fn, float fnp1) {
    float d  = fnp1 - fn;
    float th = (d == 0.0f) ? 0.0f : (fn - fnm1) / d;
    float a  = fabsf(th);
    return (a + th) / (1.0f + a);
}

__global__ __launch_bounds__(TPB) void
pbe_hires_kernel(const float* __restrict__ f0,
                 const float* __restrict__ c,
                 const float* __restrict__ g_k1_p,
                 const float* __restrict__ g_k2_p,
                 float* __restrict__ out)
{
    __shared__ float sf[2][N_MAX];  // 24 KB ping-pong CSD buffers (LDS-resident scan)
    __shared__ float sc[M_MAX];     // 1.6 KB: c trajectory -> precomputed growth scalars

    const int b   = blockIdx.x;
    const int tid = threadIdx.x;

    // ---- Stage inputs into LDS via CDNA5 async data movers (ASYNCcnt) ----
    if (tid < N_MAX / 4) {          // 750 x b128 covers 3000 floats -> sf[0]
        async_load_b128(lds_off(&sf[0][tid * 4]), f0 + (size_t)b * N_MAX + tid * 4);
    }
    if (tid < M_MAX / 4) {          // 100 x b128 covers 400 floats
        async_load_b128(lds_off(&sc[tid * 4]), c + (size_t)b * M_MAX + tid * 4);
    }
    wait_async0();
    __syncthreads();

    const float gk1 = *g_k1_p;
    const float gk2 = *g_k2_p;

    // ---- Precompute per-step growth scalar s[m] = gk1*((c-50)/50)^gk2 ----
    // Hoists the transcendental off the 400-step serial critical path.
    if (tid < M_MAX) {
        float x = (sc[tid] - 50.0f) / 50.0f;
        sc[tid] = gk1 * powf(x, gk2);
    }

    // Per-element constants: g[i] = s * w[i], w[i] = 1 + 0.1*l_i (wrap on i+1)
    float w[EPT], wp[EPT];
    #pragma unroll
    for (int e = 0; e < EPT; ++e) {
        int i = tid + e * TPB;
        if (i < N_MAX) {
            int ip1 = (i + 1 == N_MAX) ? 0 : i + 1;
            w[e]  = 1.0f + 0.1f * (DL * (float)i);
            wp[e] = 1.0f + 0.1f * (DL * (float)ip1);
        }
    }
    __syncthreads();

    // ---- 400 sequential time steps; ping-pong => 1 barrier per step ----
    for (int m = 0; m < M_MAX; ++m) {
        const int pb = m & 1;
        const float* __restrict__ src = sf[pb];
        float* __restrict__       dst = sf[pb ^ 1];

        const float s = sc[m];        // broadcast DS read (no conflict)

        #pragma unroll
        for (int e = 0; e < EPT; ++e) {
            int i = tid + e * TPB;
            if (i < N_MAX) {
                int im1 = (i == 0) ? N_MAX - 1 : i - 1;
                int ip1 = (i + 1 == N_MAX) ? 0 : i + 1;
                int ip2 = (ip1 + 1 == N_MAX) ? 0 : ip1 + 1;

                float fi   = src[i];
                float fim1 = src[im1];   // f_p1 in reference (roll +1)
                float fip1 = src[ip1];   // f_m1 in reference (roll -1)
                float fip2 = src[ip2];   // f_m2 in reference (roll -2)

                float g   = s * w[e];
                float gm1 = s * wp[e];

                // fl_n = limiter(f_m1, f, f_p1); fl_nm1 = limiter(f_m2, f_m1, f)
                float fl_n   = limiter(fip1, fi,   fim1);
                float fl_nm1 = limiter(fip2, fip1, fi);

                dst[i] = fi
                       - KH * (g * fi - gm1 * fip1)
                       - ( K2H * g   * (1.0f - KH * g)   * (fim1 - fi)   * fl_n
                         - K2H * gm1 * (1.0f - KH * gm1) * (fi   - fip1) * fl_nm1 );
            }
        }
        // One barrier suffices: passing it proves every wave finished reading
        // src and writing dst for this step, so next step may overwrite src.
        __syncthreads();
    }

    // M_MAX is even -> final state is in sf[0]. Drain LDS -> HBM async.
    if (tid < N_MAX / 4) {
        async_store_b128(out + (size_t)b * N_MAX + tid * 4,
                         lds_off(&sf[0][tid * 4]));
    }
    wait_async0();
}

extern "C" void kernel_launch(void* const* d_in, const int* in_sizes, int n_in,
                              void* d_out, int out_size, void* d_ws, size_t ws_size,
                              hipStream_t stream) {
    // setup_inputs() order: f0 [B,3000], c [B,400], temperature [B,400] (unused),
    //                       g_k1 (scalar), g_k2 (scalar)
    const float* f0  = (const float*)d_in[0];
    const float* c   = (const float*)d_in[1];
    const float* gk1 = (const float*)d_in[3];
    const float* gk2 = (const float*)d_in[4];
    float*       out = (float*)d_out;

    const int B = in_sizes[0] / N_MAX;   // 32

    pbe_hires_kernel<<<dim3(B), dim3(TPB), 0, stream>>>(f0, c, gk1, gk2, out);
}